// RNN_45749991637269
// MI455X (gfx1250) — compile-verified
//
#include <hip/hip_runtime.h>

typedef __attribute__((ext_vector_type(2))) float v2f;
typedef __attribute__((ext_vector_type(8))) float v8f;

#define NQ 4
#define NAangles 30
#define KDIM 784
#define ROWS_PER_WG 128   // 8 waves * 16 rows
#define WG_THREADS 256

__global__ __launch_bounds__(WG_THREADS, 2)
void qcirc_fused_kernel(const float* __restrict__ in,
                        const float* __restrict__ W,
                        const float* __restrict__ bias,
                        float* __restrict__ out,
                        int Btot)
{
    __shared__ float lds_ang[ROWS_PER_WG][33];   // padded stride vs 32 to avoid bank conflicts

    const int tid  = threadIdx.x;
    const int wave = tid >> 5;
    const int lane = tid & 31;

    // ---------------- Phase 1: GEMM angles = in @ W^T via WMMA f32 16x16x4 ----------------
    // A fragment layout (ISA 7.12.2, 32-bit A 16x4): lane = M (mod 16);
    //   lanes 0-15 hold K={0,1} in VGPR0/1, lanes 16-31 hold K={2,3}.
    // B fragment (4x16, row striped across lanes): lane = N (mod 16);
    //   lane-group 0 holds rows K={0,1}, group 1 holds K={2,3}.
    const int m      = lane & 15;
    const int khalf  = lane >> 4;              // 0 or 1
    const int rowTile = blockIdx.x * ROWS_PER_WG + wave * 16;

    int arow = rowTile + m;
    if (arow >= Btot) arow = Btot - 1;         // clamp (B is a multiple of 128 anyway)
    const float* aptr = in + (size_t)arow * KDIM + 2 * khalf;

    const int n0 = lane & 15;
    const int n1 = 16 + n0;
    // Lanes with n1 >= 30 read W row 0 instead: they produce a duplicate of
    // column 0 in padding columns 30/31 of the angle tile, which the circuit
    // phase never reads -- finite garbage, no mask multiply needed in the
    // hot loop.
    const float* wptr0 = W + (size_t)n0 * KDIM + 2 * khalf;
    const float* wptr1 = W + (size_t)((n1 < NAangles) ? n1 : 0) * KDIM + 2 * khalf;

    v8f c0 = {};
    v8f c1 = {};
#pragma unroll 4
    for (int k0 = 0; k0 < KDIM; k0 += 4) {
        v2f a  = *(const v2f*)(aptr  + k0);
        v2f b0 = *(const v2f*)(wptr0 + k0);
        v2f b1 = *(const v2f*)(wptr1 + k0);
        c0 = __builtin_amdgcn_wmma_f32_16x16x4_f32(false, a, false, b0, (short)0, c0, false, false);
        c1 = __builtin_amdgcn_wmma_f32_16x16x4_f32(false, a, false, b1, (short)0, c1, false, false);
    }

    // C/D layout: VGPR v -> row (v + 8*laneGroup), lane&15 -> column
    {
        const int rOff = wave * 16 + khalf * 8;
        const int col  = lane & 15;
#pragma unroll
        for (int v = 0; v < 8; ++v) {
            lds_ang[rOff + v][col]      = c0[v];
            lds_ang[rOff + v][16 + col] = c1[v];
        }
    }
    __syncthreads();

    // ---------------- Phase 2: 4-qubit circuit, one row per thread (tid < 128) -------------
    if (tid < ROWS_PER_WG) {
        const int row = blockIdx.x * ROWS_PER_WG + tid;
        if (row < Btot) {
            float ang[NAangles];
#pragma unroll
            for (int j = 0; j < NAangles; ++j)
                ang[j] = lds_ang[tid][j] + bias[j];   // bias read is wave-uniform -> s_load

            float sr[16], si[16];
#pragma unroll
            for (int i = 0; i < 16; ++i) { sr[i] = 0.0f; si[i] = 0.0f; }
            sr[0] = 1.0f;

            // RXX on qubits (w, w+1): new[i] = c*old[i] + (-i s)*old[i ^ (3 << (2-w))]
            auto rxx = [&](int w, float theta) {
                float s, c;
                __sincosf(0.5f * theta, &s, &c);
                const int mk = 3 << (2 - w);
#pragma unroll
                for (int i = 0; i < 16; ++i) {
                    const int j = i ^ mk;
                    if (j > i) {
                        float ar = sr[i], ai = si[i], br = sr[j], bi = si[j];
                        sr[i] = c * ar + s * bi;   si[i] = c * ai - s * br;
                        sr[j] = c * br + s * ai;   si[j] = c * bi - s * ar;
                    }
                }
            };

            // U3 on qubit w: [[c, -e^{i l} s], [e^{i p} s, e^{i(p+l)} c]]
            auto u3 = [&](int w, float th, float ph, float la) {
                float sh, chv;  __sincosf(0.5f * th, &sh, &chv);
                float sl, cl;   __sincosf(la, &sl, &cl);
                float sp, cp;   __sincosf(ph, &sp, &cp);
                const float u01r = -cl * sh,           u01i = -sl * sh;
                const float u10r =  cp * sh,           u10i =  sp * sh;
                const float plr  = cp * cl - sp * sl,  pli  = cp * sl + sp * cl;
                const float u11r = plr * chv,          u11i = pli * chv;
                const int mk = 1 << (3 - w);
#pragma unroll
                for (int i = 0; i < 16; ++i) {
                    if ((i & mk) == 0) {
                        const int j = i | mk;
                        float ar = sr[i], ai = si[i], br = sr[j], bi = si[j];
                        sr[i] = chv * ar + u01r * br - u01i * bi;
                        si[i] = chv * ai + u01r * bi + u01i * br;
                        sr[j] = u10r * ar - u10i * ai + u11r * br - u11i * bi;
                        si[j] = u10r * ai + u10i * ar + u11r * bi + u11i * br;
                    }
                }
            };

            int k = 0;
#pragma unroll
            for (int lay = 0; lay < 2; ++lay) {
#pragma unroll
                for (int q = 0; q < NQ - 1; ++q) {
                    rxx(q, ang[k]); k += 1;
                    u3(q, ang[k], ang[k + 1], ang[k + 2]); k += 3;
                }
                u3(NQ - 1, ang[k], ang[k + 1], ang[k + 2]); k += 3;
            }

            // out[row][amp][{re,im}]
            float* o = out + (size_t)row * 32;
#pragma unroll
            for (int i = 0; i < 16; ++i) {
                o[2 * i]     = sr[i];
                o[2 * i + 1] = si[i];
            }
        }
    }
}

extern "C" void kernel_launch(void* const* d_in, const int* in_sizes, int n_in,
                              void* d_out, int out_size, void* d_ws, size_t ws_size,
                              hipStream_t stream)
{
    const float* in   = (const float*)d_in[0];   // (B, 784)
    const float* W    = (const float*)d_in[1];   // (30, 784)
    const float* bias = (const float*)d_in[2];   // (30,)
    float*       out  = (float*)d_out;           // (B, 16, 2)

    const int Btot = in_sizes[0] / KDIM;
    const int grid = (Btot + ROWS_PER_WG - 1) / ROWS_PER_WG;

    qcirc_fused_kernel<<<grid, WG_THREADS, 0, stream>>>(in, W, bias, out, Btot);
}